// Model_41266045780624
// MI455X (gfx1250) — compile-verified
//
#include <hip/hip_runtime.h>
#include <math.h>

// ---------------------------------------------------------------------------
// Problem: Y = sigmoid( mean_{H,W}( relu(X*scale + shift) ) @ W^T + bias )
//   X: (64, 256, 128, 128) f32   W: (512, 256) f32   out: (64, 512) f32
//
// Stage 1: streaming BN+ReLU+global-avg-pool.  1 GiB read @ 23.3 TB/s ~ 46us,
//          totally bandwidth bound -> b128 non-temporal loads (no reuse, 1 GiB
//          stream >> 192 MB L2), wave32 shuffle reduction.  One block per
//          (b,c) plane (64 KB contiguous).
// Stage 2: tiny f32 GEMM (64x256 @ 256x512) with V_WMMA_F32_16X16X4_F32
//          (full f32 precision, matches reference), fused bias + sigmoid
//          epilogue using v_rcp_f32.  One wave per 16x16 output tile.
// ---------------------------------------------------------------------------

typedef float v2f __attribute__((ext_vector_type(2)));
typedef float v4f __attribute__((ext_vector_type(4)));
typedef float v8f __attribute__((ext_vector_type(8)));

#define C_DIM   256
#define HW      (128 * 128)          // 16384 elements per plane
#define OUT_DIM 512
#define B_DIM   64
#define BN_EPS  1e-5f

// ---------------------------------------------------------------------------
// Kernel 1: per-plane fused BN(eval) + ReLU + mean over HxW.
// grid = B*C blocks, 256 threads (8 wave32s).  Each thread: 16 x b128 NT
// loads, unroll 8 -> 8 loads in flight per thread.
// ---------------------------------------------------------------------------
__global__ void __launch_bounds__(256)
bn_relu_pool_kernel(const float* __restrict__ X,
                    const float* __restrict__ gamma,
                    const float* __restrict__ beta,
                    const float* __restrict__ rmean,
                    const float* __restrict__ rvar,
                    float* __restrict__ pooled /* (B, C) */) {
    const int plane = blockIdx.x;               // b * C + c
    const int c     = plane & (C_DIM - 1);

    const float s = gamma[c] * rsqrtf(rvar[c] + BN_EPS);
    const float t = fmaf(-s, rmean[c], beta[c]);   // beta - s*mean

    const v4f* __restrict__ p =
        reinterpret_cast<const v4f*>(X + (size_t)plane * HW);

    float sum = 0.0f;
    // HW/4 = 4096 float4 chunks, 256 threads -> 16 iterations, fully coalesced.
    #pragma unroll 8
    for (int i = threadIdx.x; i < HW / 4; i += 256) {
        v4f v = __builtin_nontemporal_load(&p[i]);   // streaming: NT hint
        sum += fmaxf(fmaf(v.x, s, t), 0.0f);
        sum += fmaxf(fmaf(v.y, s, t), 0.0f);
        sum += fmaxf(fmaf(v.z, s, t), 0.0f);
        sum += fmaxf(fmaf(v.w, s, t), 0.0f);
    }

    // wave32 butterfly reduction
    #pragma unroll
    for (int off = 16; off > 0; off >>= 1)
        sum += __shfl_down(sum, off, 32);

    __shared__ float red[8];
    const int lane = threadIdx.x & 31;
    const int wid  = threadIdx.x >> 5;
    if (lane == 0) red[wid] = sum;
    __syncthreads();

    if (threadIdx.x < 8) {
        float v = red[threadIdx.x];
        #pragma unroll
        for (int off = 4; off > 0; off >>= 1)
            v += __shfl_down(v, off, 32);
        if (threadIdx.x == 0)
            pooled[plane] = v * (1.0f / (float)HW);
    }
}

// ---------------------------------------------------------------------------
// Kernel 2: logits = pooled @ W^T + bias; out = sigmoid(logits).
// One wave per 16x16 output tile; K-loop of 64 x V_WMMA_F32_16X16X4_F32.
//
// A (16x4 f32) per-lane fragment: M = lane%16, K = 2*(lane/16) + {0,1}
// B (4x16 f32) per-lane fragment: N = lane%16, K = 2*(lane/16) + {0,1}
//   (B[k][n] = W[n][k], contiguous in k -> float2 load from W row n)
// C/D (16x16 f32): vgpr v, lane L -> M = v + 8*(L/16), N = L%16
// ---------------------------------------------------------------------------
__global__ void __launch_bounds__(32)
wmma_proj_sigmoid_kernel(const float* __restrict__ pooled, /* (B, C)   */
                         const float* __restrict__ W,      /* (OUT, C) */
                         const float* __restrict__ bias,   /* (OUT,)   */
                         float* __restrict__ out /* (B, OUT) */) {
    const int tile  = blockIdx.x;            // 0 .. (B/16)*(OUT/16)-1 = 127
    const int mBase = (tile >> 5) << 4;      // 4 tiles along batch
    const int nBase = (tile & 31) << 4;      // 32 tiles along OUT

    const int lane = threadIdx.x;            // 0..31, EXEC all ones
    const int half = lane >> 4;              // 0 or 1
    const int l16  = lane & 15;

    const float* __restrict__ aRow = pooled + (size_t)(mBase + l16) * C_DIM;
    const float* __restrict__ bRow = W      + (size_t)(nBase + l16) * C_DIM;

    v8f acc = {};
    #pragma unroll 8
    for (int k = 0; k < C_DIM; k += 4) {
        const int kk = k + half * 2;
        v2f a = *(const v2f*)(aRow + kk);
        v2f b = *(const v2f*)(bRow + kk);
        acc = __builtin_amdgcn_wmma_f32_16x16x4_f32(
            /*neg_a=*/false, a, /*neg_b=*/false, b,
            /*c_mod=*/(short)0, acc, /*reuse_a=*/false, /*reuse_b=*/false);
    }

    const float bv = bias[nBase + l16];
    #pragma unroll
    for (int v = 0; v < 8; ++v) {
        const int row = mBase + v + 8 * half;
        const float x = acc[v] + bv;
        // sigmoid via v_exp_f32 + v_rcp_f32 (avoid full IEEE divide expansion)
        const float y = __builtin_amdgcn_rcpf(1.0f + __expf(-x));
        out[(size_t)row * OUT_DIM + nBase + l16] = y;
    }
}

// ---------------------------------------------------------------------------
extern "C" void kernel_launch(void* const* d_in, const int* in_sizes, int n_in,
                              void* d_out, int out_size, void* d_ws, size_t ws_size,
                              hipStream_t stream) {
    const float* X     = (const float*)d_in[0];
    const float* gamma = (const float*)d_in[1];
    const float* beta  = (const float*)d_in[2];
    const float* rmean = (const float*)d_in[3];
    const float* rvar  = (const float*)d_in[4];
    const float* W     = (const float*)d_in[5];
    const float* bias  = (const float*)d_in[6];
    float* out    = (float*)d_out;
    float* pooled = (float*)d_ws;            // 64*256 floats = 64 KB scratch

    bn_relu_pool_kernel<<<B_DIM * C_DIM, 256, 0, stream>>>(
        X, gamma, beta, rmean, rvar, pooled);

    wmma_proj_sigmoid_kernel<<<(B_DIM / 16) * (OUT_DIM / 16), 32, 0, stream>>>(
        pooled, W, bias, out);
}